// MixHop_Model_52690658787914
// MI455X (gfx1250) — compile-verified
//
#include <hip/hip_runtime.h>

typedef __attribute__((ext_vector_type(2))) float v2f;
typedef __attribute__((ext_vector_type(8))) float v8f;

// ---------------------------------------------------------------------------
// gcn_norm kernels
// ---------------------------------------------------------------------------
__global__ void deg_kernel(const long long* __restrict__ col,
                           float* __restrict__ deg, int E) {
  int e = blockIdx.x * blockDim.x + threadIdx.x;
  if (e < E) {
    unsafeAtomicAdd(&deg[(int)col[e]], 1.0f);
  }
}

__global__ void dis_kernel(const float* __restrict__ deg,
                           float* __restrict__ dis, int n) {
  int i = blockIdx.x * blockDim.x + threadIdx.x;
  if (i < n) {
    float d = deg[i];
    dis[i] = (d > 0.0f) ? rsqrtf(fmaxf(d, 1.0f)) : 0.0f;
  }
}

__global__ void edgew_kernel(const long long* __restrict__ row,
                             const long long* __restrict__ col,
                             const float* __restrict__ dis,
                             float* __restrict__ w, int E) {
  int e = blockIdx.x * blockDim.x + threadIdx.x;
  if (e < E) {
    w[e] = dis[(int)row[e]] * dis[(int)col[e]];
  }
}

// ---------------------------------------------------------------------------
// Dense GEMM, LDS-staged weights, fused bias.
//   Y[m][n] = sum_k X[m][k] * W[k][n] + b[n]
// Block = 256 threads = 8 waves. The whole W (KIN x KOUT, <=96KB; WGP has
// 320KB LDS) is packed into LDS once per block as K-pair float2s:
//   sW[kp*KOUT + n] = { W[2kp][n], W[2kp+1][n] }
// so each WMMA B-fragment is one ds_load_b64 (pairs collapse to
// ds_load_2addr_b64). Each wave computes a full 16 x KOUT strip: one
// global_load_b64 of A per K-step, reused across KOUT/16 WMMAs.
// Per K-step, ALL B fragments are hoisted into registers before the WMMA
// burst so the ds_loads issue as a clause and s_wait_dscnt drains
// incrementally instead of stalling every WMMA pair.
//
// Fragment layouts (wave32):
//   A 16x4 f32: lanes 0-15 -> M=0..15 {K=k,k+1}; lanes 16-31 -> {K=k+2,k+3}
//   B 4x16 f32: lanes 0-15 -> N=0..15 {K=k,k+1}; lanes 16-31 -> {K=k+2,k+3}
//   C/D 16x16 f32: 8 VGPRs; lanes 0-15 -> M=r, lanes 16-31 -> M=8+r; N=lane&15
// ---------------------------------------------------------------------------
template <int KIN, int KOUT>
__global__ __launch_bounds__(256)
void wmma_gemm_lds(const float* __restrict__ X, int ldX,
                   const float* __restrict__ W,
                   const float* __restrict__ bias,
                   float* __restrict__ Y, int ldY,
                   int Mrows) {
  extern __shared__ char smem_raw[];
  v2f* sW = (v2f*)smem_raw;                       // (KIN/2) * KOUT float2s

  const int tid = threadIdx.x;

  // ---- cooperative fill: pack W K-pairs into LDS ----
  constexpr int PAIRS = (KIN / 2) * KOUT;
  for (int i = tid; i < PAIRS; i += 256) {
    const int kp = i / KOUT;
    const int n  = i - kp * KOUT;
    v2f v;
    v[0] = W[(size_t)(2 * kp) * KOUT + n];
    v[1] = W[(size_t)(2 * kp + 1) * KOUT + n];
    sW[i] = v;
  }
  __syncthreads();

  // ---- per-wave 16 x KOUT strip ----
  const int lane   = tid & 31;
  const int waveId = tid >> 5;
  const int strip  = blockIdx.x * 8 + waveId;
  if (strip * 16 >= Mrows) return;                // wave-uniform: EXEC all-ones

  const int half = lane >> 4;                     // K sub-pair selector
  const int l    = lane & 15;

  constexpr int NT = KOUT / 16;
  v8f acc[NT];
#pragma unroll
  for (int t = 0; t < NT; ++t) {
    const float bv = bias[t * 16 + l];
#pragma unroll
    for (int r = 0; r < 8; ++r) acc[t][r] = bv;
  }

  const float* xrow = X + (size_t)(strip * 16 + l) * ldX + 2 * half;

  for (int k = 0; k < KIN; k += 4) {
    const v2f a = *(const v2f*)(xrow + k);        // A fragment (global, b64)
    __builtin_prefetch(xrow + k + 64, 0, 3);      // global_prefetch_b8, A stream

    const v2f* wrow = sW + (size_t)(k / 2 + half) * KOUT + l;
    v2f bfrag[NT];                                // hoisted B fragments (LDS)
#pragma unroll
    for (int t = 0; t < NT; ++t) bfrag[t] = wrow[t * 16];
#pragma unroll
    for (int t = 0; t < NT; ++t) {
      acc[t] = __builtin_amdgcn_wmma_f32_16x16x4_f32(
          /*neg_a=*/false, a, /*neg_b=*/false, bfrag[t],
          /*c_mod=*/(short)0, acc[t], /*reuse_a=*/false, /*reuse_b=*/false);
    }
  }

  float* ybase = Y + (size_t)(strip * 16 + half * 8) * ldY + l;
#pragma unroll
  for (int t = 0; t < NT; ++t) {
#pragma unroll
    for (int r = 0; r < 8; ++r) {
      ybase[(size_t)r * ldY + t * 16] = acc[t][r];
    }
  }
}

// ---------------------------------------------------------------------------
// SpMM: O[col[e]][f] += w[e] * H[row[e]][f]   (one wave per edge)
// Edge metadata is wave-uniform -> force into SGPRs via readfirstlane so the
// feature loop is scalar-base + lane-offset vector ops. L2-resident working
// set; hardware global_atomic_add_f32 for the scatter.
// ---------------------------------------------------------------------------
__global__ __launch_bounds__(256)
void spmm_atomic(const long long* __restrict__ rows,
                 const long long* __restrict__ cols,
                 const float* __restrict__ w,
                 const float* __restrict__ H, int ldH,
                 float* __restrict__ O, int ldO,
                 int E, int F) {
  const int gid  = blockIdx.x * blockDim.x + threadIdx.x;
  const int e    = gid >> 5;
  const int lane = gid & 31;
  if (e >= E) return;                              // wave-uniform exit

  int   r  = (int)rows[e];
  int   cI = (int)cols[e];
  float wt = w[e];
  r  = __builtin_amdgcn_readfirstlane(r);
  cI = __builtin_amdgcn_readfirstlane(cI);
  wt = __uint_as_float(__builtin_amdgcn_readfirstlane(__float_as_uint(wt)));

  const float* hrow = H + (size_t)r * ldH;
  float*       orow = O + (size_t)cI * ldO;
  for (int f = lane; f < F; f += 32) {
    unsafeAtomicAdd(&orow[f], wt * hrow[f]);
  }
}

__global__ void relu_kernel(float* __restrict__ h, size_t n) {
  size_t i = (size_t)blockIdx.x * blockDim.x + threadIdx.x;
  if (i < n) h[i] = fmaxf(h[i], 0.0f);
}

// ---------------------------------------------------------------------------
// Host launch
// ---------------------------------------------------------------------------
template <int KIN, int KOUT>
static inline void launch_gemm(const float* X, int ldX, const float* W,
                               const float* b, float* Y, int ldY,
                               int M, hipStream_t s) {
  const int strips = (M + 15) / 16;
  const int blocks = (strips + 7) / 8;
  const size_t shbytes = (size_t)KIN * KOUT * sizeof(float);
  wmma_gemm_lds<KIN, KOUT><<<blocks, 256, shbytes, s>>>(X, ldX, W, b, Y, ldY, M);
}

static inline void launch_spmm(const long long* row, const long long* col,
                               const float* w, const float* H, int ldH,
                               float* O, int ldO, int E, int F, hipStream_t s) {
  const long long threads = (long long)E * 32;
  const int blocks = (int)((threads + 255) / 256);
  spmm_atomic<<<blocks, 256, 0, s>>>(row, col, w, H, ldH, O, ldO, E, F);
}

extern "C" void kernel_launch(void* const* d_in, const int* in_sizes, int n_in,
                              void* d_out, int out_size, void* d_ws, size_t ws_size,
                              hipStream_t stream) {
  const int N = 100000;
  const int E = 3200000;

  const float*     x   = (const float*)d_in[0];
  const long long* ei  = (const long long*)d_in[1];
  const float*     W1  = (const float*)d_in[2];   // [3,128,128]
  const float*     b1  = (const float*)d_in[3];   // [3,128]
  const float*     W2  = (const float*)d_in[4];   // [3,384,64]
  const float*     b2  = (const float*)d_in[5];   // [3,64]
  const float*     Wf  = (const float*)d_in[6];   // [192,64]
  const float*     bf  = (const float*)d_in[7];   // [64]
  float*           out = (float*)d_out;           // [N,64]

  const long long* row = ei;       // edge_index[0]
  const long long* col = ei + E;   // edge_index[1]

  // workspace carve-up
  float* p     = (float*)d_ws;
  float* deg   = p; p += N;
  float* dis   = p; p += N;
  float* ew    = p; p += E;
  float* tmpA  = p; p += (size_t)N * 128;   // reused as [N,64] in layer 2
  float* tmpB  = p; p += (size_t)N * 128;   // reused as [N,64] in layer 2
  float* hcat1 = p; p += (size_t)N * 384;
  float* hcat2 = p; p += (size_t)N * 192;

  // ---- gcn_norm ----
  hipMemsetAsync(deg, 0, (size_t)N * sizeof(float), stream);
  deg_kernel<<<(E + 255) / 256, 256, 0, stream>>>(col, deg, E);
  dis_kernel<<<(N + 255) / 256, 256, 0, stream>>>(deg, dis, N);
  edgew_kernel<<<(E + 255) / 256, 256, 0, stream>>>(row, col, dis, ew, E);

  // ---- layer 1: MixHop(128 -> 3*128) ----
  hipMemsetAsync(hcat1, 0, (size_t)N * 384 * sizeof(float), stream);
  // hop 0: x @ W1[0] + b1[0] -> hcat1[:, 0:128]
  launch_gemm<128, 128>(x, 128, W1, b1, hcat1, 384, N, stream);
  // hop 1: spmm(x @ W1[1] + b1[1]) -> hcat1[:, 128:256]
  launch_gemm<128, 128>(x, 128, W1 + 128 * 128, b1 + 128, tmpA, 128, N, stream);
  launch_spmm(row, col, ew, tmpA, 128, hcat1 + 128, 384, E, 128, stream);
  // hop 2: spmm(spmm(x @ W1[2] + b1[2])) -> hcat1[:, 256:384]
  launch_gemm<128, 128>(x, 128, W1 + 2 * 128 * 128, b1 + 2 * 128, tmpA, 128, N, stream);
  hipMemsetAsync(tmpB, 0, (size_t)N * 128 * sizeof(float), stream);
  launch_spmm(row, col, ew, tmpA, 128, tmpB, 128, E, 128, stream);
  launch_spmm(row, col, ew, tmpB, 128, hcat1 + 256, 384, E, 128, stream);
  // relu
  {
    const size_t n = (size_t)N * 384;
    relu_kernel<<<(unsigned)((n + 255) / 256), 256, 0, stream>>>(hcat1, n);
  }

  // ---- layer 2: MixHop(384 -> 3*64) ----
  hipMemsetAsync(hcat2, 0, (size_t)N * 192 * sizeof(float), stream);
  // hop 0
  launch_gemm<384, 64>(hcat1, 384, W2, b2, hcat2, 192, N, stream);
  // hop 1
  launch_gemm<384, 64>(hcat1, 384, W2 + 384 * 64, b2 + 64, tmpA, 64, N, stream);
  launch_spmm(row, col, ew, tmpA, 64, hcat2 + 64, 192, E, 64, stream);
  // hop 2
  launch_gemm<384, 64>(hcat1, 384, W2 + 2 * 384 * 64, b2 + 2 * 64, tmpA, 64, N, stream);
  hipMemsetAsync(tmpB, 0, (size_t)N * 64 * sizeof(float), stream);
  launch_spmm(row, col, ew, tmpA, 64, tmpB, 64, E, 64, stream);
  launch_spmm(row, col, ew, tmpB, 64, hcat2 + 128, 192, E, 64, stream);

  // ---- final linear: hcat2 @ Wf + bf -> out [N,64] ----
  launch_gemm<192, 64>(hcat2, 192, Wf, bf, out, 64, N, stream);
}